// DepthMemoryCache_34110630265266
// MI455X (gfx1250) — compile-verified
//
#include <hip/hip_runtime.h>

#define D_ 8
#define B_ 4
#define S_ 4096
#define C_ 1024
#define G_ 64

typedef float v2f __attribute__((ext_vector_type(2)));
typedef float v8f __attribute__((ext_vector_type(8)));

// ---------------------------------------------------------------------------
// Kernel 1: raw sums over S (means * S). S split into 32 chunks of 128.
// D*B*32*C threads, c fastest for coalescing. msum must be pre-zeroed.
// ---------------------------------------------------------------------------
__global__ void k_means(const float* __restrict__ cs, float* __restrict__ msum) {
    int idx   = blockIdx.x * blockDim.x + threadIdx.x;
    int c     = idx % C_;
    int rest  = idx / C_;
    int chunk = rest % 32;
    int db    = rest / 32;                     // d*B + b, 0..31
    const float* p = cs + ((size_t)db * S_ + (size_t)chunk * 128) * C_ + c;
    float sum = 0.f;
    #pragma unroll 8
    for (int s = 0; s < 128; ++s) sum += p[(size_t)s * C_];
    atomicAdd(&msum[db * C_ + c], sum);
}

// ---------------------------------------------------------------------------
// Kernel 2: keys[d][b][g] = (1/S) * sum_c msum[d][b][c] * W_u[g][c]
// One block per (d,b), one thread per g. Tiny.
// ---------------------------------------------------------------------------
__global__ void k_keys(const float* __restrict__ msum, const float* __restrict__ Wu,
                       float* __restrict__ keys) {
    int db = blockIdx.x;                       // 0..31
    int g  = threadIdx.x;                      // 0..63
    const float* m = msum + (size_t)db * C_;
    const float* w = Wu   + (size_t)g  * C_;
    float acc = 0.f;
    #pragma unroll 4
    for (int c = 0; c < C_; ++c) acc += m[c] * w[c];
    keys[db * G_ + g] = acc * (1.0f / (float)S_);
}

// ---------------------------------------------------------------------------
// Kernel 3 (WMMA): u[b][s][g] = sum_c cached[D-1][b][s][c] * W_u[g][c]
// One wave per 16-row s-tile; 4 v8f accumulators cover all G=64.
// A (16x4 f32): lane m=lane&15 holds row m; VGPR0/1 = K = (lane>>4)*2 + {0,1}
// B (4x16 f32): lane n=lane&15 holds col n; VGPR0/1 = K = (lane>>4)*2 + {0,1}
//   -> B[k][n] = W_u[g0+n][k], so same float2 addressing pattern as A.
// D (16x16):   lane n=lane&15, VGPR r holds M = (lane>>4)*8 + r.
// ---------------------------------------------------------------------------
__global__ void __launch_bounds__(32) k_proj(const float* __restrict__ cs,
                                             const float* __restrict__ Wu,
                                             float* __restrict__ u) {
    int w    = blockIdx.x;                     // 0 .. B*S/16 - 1
    int b    = w / (S_ / 16);
    int s0   = (w % (S_ / 16)) * 16;
    int lane = threadIdx.x;
    int m    = lane & 15;
    int koff = (lane >> 4) * 2;

    const float* A  = cs + (((size_t)(D_ - 1) * B_ + b) * S_ + (s0 + m)) * C_ + koff;
    const float* B0 = Wu + (size_t)m * C_ + koff;

    v8f acc0 = {}, acc1 = {}, acc2 = {}, acc3 = {};
    #pragma unroll 4
    for (int k0 = 0; k0 < C_; k0 += 4) {
        v2f a  = *(const v2f*)(A  + k0);
        v2f b0 = *(const v2f*)(B0 + k0);
        v2f b1 = *(const v2f*)(B0 + 16 * C_ + k0);
        v2f b2 = *(const v2f*)(B0 + 32 * C_ + k0);
        v2f b3 = *(const v2f*)(B0 + 48 * C_ + k0);
        acc0 = __builtin_amdgcn_wmma_f32_16x16x4_f32(false, a, false, b0, (short)0, acc0, false, false);
        acc1 = __builtin_amdgcn_wmma_f32_16x16x4_f32(false, a, false, b1, (short)0, acc1, false, false);
        acc2 = __builtin_amdgcn_wmma_f32_16x16x4_f32(false, a, false, b2, (short)0, acc2, false, false);
        acc3 = __builtin_amdgcn_wmma_f32_16x16x4_f32(false, a, false, b3, (short)0, acc3, false, false);
    }

    int n  = lane & 15;
    int mb = (lane >> 4) * 8;
    float* uo = u + ((size_t)b * S_ + s0 + mb) * G_ + n;
    #pragma unroll
    for (int r = 0; r < 8; ++r) {
        uo[(size_t)r * G_ +  0] = acc0[r];
        uo[(size_t)r * G_ + 16] = acc1[r];
        uo[(size_t)r * G_ + 32] = acc2[r];
        uo[(size_t)r * G_ + 48] = acc3[r];
    }
}

// ---------------------------------------------------------------------------
// Kernel 4: per (b,s): logits[d] = u[b,s,:] . keys[d,b,:]; softmax over d;
// out[b,s,c] = sum_d gates[d] * cached[d,b,s,c]. One 256-thread block per (b,s).
// ---------------------------------------------------------------------------
__global__ void __launch_bounds__(256) k_out(const float* __restrict__ cs,
                                             const float* __restrict__ u,
                                             const float* __restrict__ keys,
                                             float* __restrict__ out) {
    int bs   = blockIdx.x;                     // b*S + s
    int b    = bs / S_;
    int s    = bs % S_;
    int t    = threadIdx.x;
    int wv   = t >> 5;                         // wave id == d (0..7)
    int lane = t & 31;

    __shared__ float sm[D_];
    const float* ub = u + (size_t)bs * G_;
    {
        const float* kb = keys + (wv * B_ + b) * G_;
        float p = ub[lane] * kb[lane] + ub[lane + 32] * kb[lane + 32];
        #pragma unroll
        for (int off = 16; off > 0; off >>= 1)
            p += __shfl_xor(p, off, 32);
        if (lane == 0) sm[wv] = p;
    }
    __syncthreads();

    float mx = sm[0];
    #pragma unroll
    for (int d = 1; d < D_; ++d) mx = fmaxf(mx, sm[d]);
    float gate[D_];
    float denom = 0.f;
    #pragma unroll
    for (int d = 0; d < D_; ++d) { gate[d] = __expf(sm[d] - mx); denom += gate[d]; }
    float inv = 1.0f / denom;

    // weighted combine, 4 floats per thread (float4 loads, 512B per wave/d)
    float4 acc = make_float4(0.f, 0.f, 0.f, 0.f);
    #pragma unroll
    for (int d = 0; d < D_; ++d) {
        const float4 v = *(const float4*)(cs + (((size_t)d * B_ + b) * S_ + s) * C_ + 4 * t);
        float g = gate[d] * inv;
        acc.x += g * v.x; acc.y += g * v.y; acc.z += g * v.z; acc.w += g * v.w;
    }
    *(float4*)(out + (size_t)bs * C_ + 4 * t) = acc;
}

// ---------------------------------------------------------------------------
extern "C" void kernel_launch(void* const* d_in, const int* in_sizes, int n_in,
                              void* d_out, int out_size, void* d_ws, size_t ws_size,
                              hipStream_t stream) {
    const float* cs = (const float*)d_in[0];   // [D,B,S,C]
    const float* Wu = (const float*)d_in[1];   // [G,C]
    float* out = (float*)d_out;                // [B,S,C]

    char*  ws   = (char*)d_ws;
    float* msum = (float*)ws;                            // D*B*C   = 128 KB
    float* keys = (float*)(ws + (size_t)D_ * B_ * C_ * 4);          // 8 KB
    float* u    = (float*)(ws + (size_t)D_ * B_ * C_ * 4
                              + (size_t)D_ * B_ * G_ * 4);          // 4 MB

    hipMemsetAsync(msum, 0, (size_t)D_ * B_ * C_ * sizeof(float), stream);

    // mean reduction: D*B*32*C threads
    k_means<<<(D_ * B_ * 32 * C_) / 256, 256, 0, stream>>>(cs, msum);

    // WMMA projection of last depth: one wave per 16 s-rows
    k_proj<<<(B_ * S_) / 16, 32, 0, stream>>>(cs, Wu, u);

    // keys from means
    k_keys<<<D_ * B_, G_, 0, stream>>>(msum, Wu, keys);

    // softmax gates + weighted combine
    k_out<<<B_ * S_, 256, 0, stream>>>(cs, u, keys, out);
}